// BasePointPWL_11184094839093
// MI455X (gfx1250) — compile-verified
//
#include <hip/hip_runtime.h>

// BasePointPWL calibrate1d: out[n,c] = PWL interp of x[n,c] against per-channel
// sorted breakpoints xp[c,:] and values yp[c,:]. N=1048576, C=64, K=64.
// Memory-bound: 512 MB HBM traffic -> ~22 us floor @ 23.3 TB/s.
// CDNA5 path: async global->LDS staging (ASYNCcnt + s_wait_asynccnt), b128 streaming.
// Hot loop: affine guess + exact searchsorted verify (2 ds_load_b64 + 1 fma per
// element, zero iterations of the verify loops on uniform tables), no divides.
// Fast path has no bounds checks: constant global offsets fold into IOFFSET.

#define CCH   64
#define KK    64
#define KPAD  65           // +1 entry row pad to rotate LDS banks per channel
#define BLOCK 256
#define ELEMS 32           // float4 elements per thread

__global__ __launch_bounds__(BLOCK) void
pwl_calibrate_kernel(const float* __restrict__ x,
                     const float* __restrict__ xp,
                     const float* __restrict__ yp,
                     float* __restrict__ out,
                     long long total4)
{
    __shared__ float2 s_A[CCH * KPAD];   // {xp[k], yp[k]}
    __shared__ float2 s_B[CCH * KPAD];   // {xp[k], slope[k]}, slope[k] over segment (k-1,k)

    const int tid = threadIdx.x;

    // ---- Stage raw xp/yp into the s_A region with async global->LDS B128 ----
    // raw xp: floats [0..4095], raw yp: floats [4096..8191] of s_A (8320 floats avail).
    {
        unsigned ldsA = (unsigned)(size_t)(void*)s_A;
        #pragma unroll
        for (int r = 0; r < (CCH * KK / 4) / BLOCK; ++r) {   // 4 b128 chunks/thread/table
            int g = tid + r * BLOCK;                         // chunk id 0..1023
            unsigned goff = (unsigned)(g << 4);
            unsigned lx   = ldsA + goff;                     // raw xp
            unsigned ly   = ldsA + 16384u + goff;            // raw yp
            asm volatile("global_load_async_to_lds_b128 %0, %1, %2"
                         :: "v"(lx), "v"(goff), "s"(xp) : "memory");
            asm volatile("global_load_async_to_lds_b128 %0, %1, %2"
                         :: "v"(ly), "v"(goff), "s"(yp) : "memory");
        }
        asm volatile("s_wait_asynccnt 0" ::: "memory");
    }
    __syncthreads();

    // ---- Build interleaved tables. 4096 entries / 256 threads = 16 each. ----
    // Pass 1: build s_B (disjoint from raw region) and stash s_A entries in regs.
    const float* rawx = (const float*)s_A;                // [4096]
    const float* rawy = ((const float*)s_A) + CCH * KK;   // [4096]
    float2 areg[(CCH * KK) / BLOCK];
    #pragma unroll
    for (int r = 0; r < (CCH * KK) / BLOCK; ++r) {
        int e  = tid + r * BLOCK;          // 0..4095
        int ch = e >> 6;
        int k  = e & (KK - 1);
        int km = k ? k : 1;                // slot k=0 never read (i in [1,63])
        int be = (ch << 6) + km;
        float xpk = rawx[e];
        float ypk = rawy[e];
        float dx  = rawx[be] - rawx[be - 1] + 1e-7f;
        float dy  = rawy[be] - rawy[be - 1];
        s_B[ch * KPAD + k] = make_float2(xpk, dy / dx);
        areg[r] = make_float2(xpk, ypk);
    }
    __syncthreads();
    // Pass 2: overwrite s_A region with interleaved {xp, yp} entries.
    #pragma unroll
    for (int r = 0; r < (CCH * KK) / BLOCK; ++r) {
        int e  = tid + r * BLOCK;
        int ch = e >> 6;
        int k  = e & (KK - 1);
        s_A[ch * KPAD + k] = areg[r];
    }
    __syncthreads();

    // ---- Per-thread channel quad + affine guess parameters -----------------
    // flat element = i4*4 + j ; channel = flat & 63 ; with i4 = blk*8192 + it*256 + tid
    // the channel quad is fixed per thread: c0 = (tid & 15) * 4.
    const int c0 = (tid & 15) << 2;
    float x0[4], invh[4];
    #pragma unroll
    for (int j = 0; j < 4; ++j) {
        int cb = (c0 + j) * KPAD;
        float xlo = s_B[cb + 0].x;
        float xhi = s_B[cb + KK - 1].x;
        x0[j]   = xlo;
        invh[j] = (float)(KK - 1) / (xhi - xlo + 1e-12f);
    }
    const float2* A0 = s_A + c0 * KPAD;
    const float2* B0 = s_B + c0 * KPAD;

    const long long blkbase = (long long)blockIdx.x * (BLOCK * ELEMS);
    const float4* __restrict__ xb = ((const float4*)x) + blkbase + tid;
    float4*       __restrict__ ob = ((float4*)out)     + blkbase + tid;

    // One element: affine guess + exact searchsorted fix-up + fused interp.
    auto pwl1 = [&](float xv, int j) -> float {
        const float2* Aj = A0 + j * KPAD;
        const float2* Bj = B0 + j * KPAD;
        int g = 1 + __float2int_rz((xv - x0[j]) * invh[j]);
        g = (g < 1) ? 1 : g;
        g = (g > KK - 1) ? (KK - 1) : g;
        float2 a = Aj[g - 1];              // {xp[g-1], yp[g-1]}
        float2 b = Bj[g];                  // {xp[g],   slope[g]}
        // target invariant: (g==1 || xp[g-1] < xv) && (g==K-1 || xv <= xp[g])
        while (g > 1 && xv <= a.x)     { --g; a = Aj[g - 1]; b = Bj[g]; }
        while (g < KK - 1 && b.x < xv) { ++g; a = Aj[g - 1]; b = Bj[g]; }
        return __builtin_fmaf(xv - a.x, b.y, a.y);
    };

    if (blkbase + (BLOCK * ELEMS) <= total4) {
        // ---- Fast path: whole block in range, no per-iteration bounds math ----
        #pragma unroll 4
        for (int it = 0; it < ELEMS; ++it) {
            float4 v = xb[it * BLOCK];     // constant byte offsets -> IOFFSET
            float4 ov;
            ov.x = pwl1(v.x, 0);
            ov.y = pwl1(v.y, 1);
            ov.z = pwl1(v.z, 2);
            ov.w = pwl1(v.w, 3);
            ob[it * BLOCK] = ov;
        }
    } else {
        // ---- Tail path: guarded ----
        for (int it = 0; it < ELEMS; ++it) {
            long long i4 = blkbase + tid + (long long)it * BLOCK;
            if (i4 >= total4) break;
            float4 v = xb[it * BLOCK];
            float4 ov;
            ov.x = pwl1(v.x, 0);
            ov.y = pwl1(v.y, 1);
            ov.z = pwl1(v.z, 2);
            ov.w = pwl1(v.w, 3);
            ob[it * BLOCK] = ov;
        }
    }
}

extern "C" void kernel_launch(void* const* d_in, const int* in_sizes, int n_in,
                              void* d_out, int out_size, void* d_ws, size_t ws_size,
                              hipStream_t stream) {
    const float* x  = (const float*)d_in[0];   // [N, C]
    const float* xp = (const float*)d_in[1];   // [C, K]
    const float* yp = (const float*)d_in[2];   // [C, K]
    float* out = (float*)d_out;                // [N, C]

    long long total  = (long long)in_sizes[0]; // N*C (divisible by 4)
    long long total4 = total >> 2;
    long long perblk = (long long)BLOCK * ELEMS;
    int blocks = (int)((total4 + perblk - 1) / perblk);

    pwl_calibrate_kernel<<<blocks, BLOCK, 0, stream>>>(x, xp, yp, out, total4);
}